// RNTN_2929167696070
// MI455X (gfx1250) — compile-verified
//
#include <hip/hip_runtime.h>
#include <math.h>

#define H   192
#define H2  384
#define NB  8
#define NL  32
#define NNODES 504   // B*(2L-1)
#define OUTC 5

typedef __attribute__((ext_vector_type(2))) float v2f;
typedef __attribute__((ext_vector_type(8))) float v8f;

// ---------------- leaves: nodeBuf[0..255] = embed[leaf_ids] ----------------
__global__ void leaf_gather(const float* __restrict__ embed,
                            const int* __restrict__ leaf_ids,
                            float* __restrict__ nodeBuf) {
  int tid = blockIdx.x * blockDim.x + threadIdx.x;
  if (tid >= NB * NL * H) return;
  int row = tid / H;
  int f   = tid - row * H;
  nodeBuf[tid] = embed[(size_t)leaf_ids[row] * H + f];
}

// ------------- build C[x, j] = concat(left, right) of sibling pairs --------
// Cmat stored row-major [384][Npad]; columns >= N zero-filled (WMMA padding).
__global__ void build_C(const float* __restrict__ nodeBuf,
                        float* __restrict__ Cmat,
                        int N, int Npad, int prevBase, int nPrev) {
  int tid = blockIdx.x * blockDim.x + threadIdx.x;
  if (tid >= H2 * Npad) return;
  int x = tid / Npad;
  int j = tid - x * Npad;
  float v = 0.f;
  if (j < N) {
    int half = nPrev >> 1;
    int b = j / half;
    int i = j - b * half;
    int childRow = prevBase + b * nPrev + 2 * i + (x >= H ? 1 : 0);
    int f = (x >= H) ? (x - H) : x;
    v = nodeBuf[(size_t)childRow * H + f];
  }
  Cmat[(size_t)x * Npad + j] = v;
}

// ------------- bilinear level: out[n,h] = tanh(c^T V_h c + W^T c + b) ------
// Block = 128 threads = 4 waves, each wave owns one h (h = blockIdx.y*4+wave).
// Block stages a 384x16 tile of C in LDS (24 KB), shared by all 4 waves.
// Per wave: loop 24 x-blocks; inner K-loop of 96 v_wmma_f32_16x16x4_f32;
// second contraction (multiply by C[x,n], sum over x) fused in-register.
__global__ __launch_bounds__(128) void bilinear_level(
    const float* __restrict__ Vt, const float* __restrict__ Cmat,
    const float* __restrict__ W,  const float* __restrict__ bvec,
    float* __restrict__ nodeBuf, int N, int Npad, int outBase) {
  __shared__ float ldsC[H2 * 16];

  int n0 = blockIdx.x * 16;
  for (int e = threadIdx.x; e < H2 * 16; e += blockDim.x) {
    int row = e >> 4;
    int nn  = e & 15;
    ldsC[e] = Cmat[(size_t)row * Npad + n0 + nn];
  }
  __syncthreads();

  int wave = threadIdx.x >> 5;
  int lane = threadIdx.x & 31;
  int h    = blockIdx.y * 4 + wave;       // 0..191
  int n    = lane & 15;                    // output column within tile
  int hi   = lane >> 4;                    // 0 or 1 (lane half)
  int yo   = 2 * hi;                       // K sub-offset for this lane half

  float acc = 0.f;
  const float* Abase = Vt + (size_t)h * H2 * H2;   // Vt[h, :, :]

  for (int xb = 0; xb < H2 / 16; ++xb) {
    // A rows: M = x-local (lanes 0-15 and 16-31 both map M=0..15)
    const float* Arow = Abase + (size_t)(xb * 16 + n) * H2 + yo;
    v8f d = {};
    for (int k = 0; k < H2; k += 4) {
      v2f a = *(const v2f*)(Arow + k);                       // K pair (k+yo, k+yo+1)
      v2f bb = { ldsC[(k + yo) * 16 + n],                    // B[K, n]
                 ldsC[(k + yo + 1) * 16 + n] };
      d = __builtin_amdgcn_wmma_f32_16x16x4_f32(
              false, a, false, bb, (short)0, d, false, false);
    }
    // Fused second contraction: sum_x D[x,n] * C[x,n] for this x-block.
    // Lane holds rows xr..xr+7 (VGPR v -> M = v + 8*hi).
    int xr = xb * 16 + 8 * hi;
    float part = 0.f;
#pragma unroll
    for (int v = 0; v < 8; ++v) part += d[v] * ldsC[(xr + v) * 16 + n];
    acc += part;
  }
  // combine the two lane halves (same n, complementary x rows)
  acc += __shfl_xor(acc, 16, 32);

  // Wx: lane-uniform W reads (scalarized), LDS reads for C
  float wx = 0.f;
  for (int x = 0; x < H2; ++x) wx += ldsC[x * 16 + n] * W[(size_t)x * H + h];

  float s = acc + wx + bvec[h];
  float t = tanhf(s);
  if (lane < 16 && (n0 + n) < N)
    nodeBuf[(size_t)(outBase + n0 + n) * H + h] = t;
}

// ------------- classifier + log_softmax: one wave per node row -------------
__global__ void classify(const float* __restrict__ nodeBuf,
                         const float* __restrict__ Ww,
                         const float* __restrict__ Wb,
                         float* __restrict__ out) {
  int r = blockIdx.x;
  int lane = threadIdx.x;   // 32 threads
  float p[OUTC] = {0.f, 0.f, 0.f, 0.f, 0.f};
  for (int f = lane; f < H; f += 32) {
    float v = nodeBuf[(size_t)r * H + f];
#pragma unroll
    for (int o = 0; o < OUTC; ++o) p[o] += v * Ww[o * H + f];
  }
#pragma unroll
  for (int o = 0; o < OUTC; ++o)
    for (int off = 16; off > 0; off >>= 1) p[o] += __shfl_xor(p[o], off, 32);

  if (lane == 0) {
    float lg[OUTC];
    float m = -3.4e38f;
#pragma unroll
    for (int o = 0; o < OUTC; ++o) { lg[o] = p[o] + Wb[o]; m = fmaxf(m, lg[o]); }
    float ssum = 0.f;
#pragma unroll
    for (int o = 0; o < OUTC; ++o) ssum += expf(lg[o] - m);
    float lse = m + logf(ssum);
#pragma unroll
    for (int o = 0; o < OUTC; ++o) out[r * OUTC + o] = lg[o] - lse;
  }
}

extern "C" void kernel_launch(void* const* d_in, const int* in_sizes, int n_in,
                              void* d_out, int out_size, void* d_ws, size_t ws_size,
                              hipStream_t stream) {
  (void)in_sizes; (void)n_in; (void)out_size; (void)ws_size;
  const float* embed    = (const float*)d_in[0];
  const float* Vt       = (const float*)d_in[1];
  const float* W        = (const float*)d_in[2];
  const float* bvec     = (const float*)d_in[3];
  const float* Ww       = (const float*)d_in[4];
  const float* Wb       = (const float*)d_in[5];
  const int*   leaf_ids = (const int*)d_in[6];
  float* out = (float*)d_out;

  float* nodeBuf = (float*)d_ws;                // 504*192 f32
  float* Cmat    = nodeBuf + NNODES * H;        // 384*128 f32 (max, padded)

  {
    int total = NB * NL * H;
    leaf_gather<<<(total + 255) / 256, 256, 0, stream>>>(embed, leaf_ids, nodeBuf);
  }

  const int Ns[5]     = {128, 64, 32, 16, 8};
  const int Npads[5]  = {128, 64, 32, 16, 16};
  const int prevB[5]  = {0, 256, 384, 448, 480};
  const int nPrevs[5] = {32, 16, 8, 4, 2};
  const int outB[5]   = {256, 384, 448, 480, 496};

  for (int l = 0; l < 5; ++l) {
    int total = H2 * Npads[l];
    build_C<<<(total + 255) / 256, 256, 0, stream>>>(
        nodeBuf, Cmat, Ns[l], Npads[l], prevB[l], nPrevs[l]);
    dim3 grid(Npads[l] / 16, 48);   // 48*4 waves = 192 h values
    bilinear_level<<<grid, 128, 0, stream>>>(
        Vt, Cmat, W, bvec, nodeBuf, Ns[l], Npads[l], outB[l]);
  }

  classify<<<NNODES, 32, 0, stream>>>(nodeBuf, Ww, Wb, out);
}